// PatchBasedConv2D_11879879541169
// MI455X (gfx1250) — compile-verified
//
#include <hip/hip_runtime.h>

typedef float v2f __attribute__((ext_vector_type(2)));
typedef float v8f __attribute__((ext_vector_type(8)));

#define TW 32            // z tile width per workgroup
#define TH 16            // z tile height per workgroup
#define CW (TW + 2)      // conv1 (y) tile width  = 34
#define CH (TH + 2)      // conv1 (y) tile height = 18
#define XW (TW + 4)      // staged x tile width   = 36
#define XH (TH + 4)      // staged x tile height  = 20
#define NPIX (CW * CH)   // conv1 pixels per tile = 612
#define NPAD 624         // NPIX rounded up to 16-pixel WMMA groups (39*16)

__global__ __launch_bounds__(256) void fused_conv2x_relu_kernel(
    const float* __restrict__ x,   // [16,512,512,3]
    const float* __restrict__ w1,  // [3,3,3,4] HWIO -> flat [t][ic][oc], t=ky*3+kx
    const float* __restrict__ b1,  // [4]
    const float* __restrict__ w2,  // [3,3,4,1] -> flat [t][c]
    const float* __restrict__ b2,  // [1]
    float* __restrict__ out)       // [16,508,508,1]
{
    __shared__ float xs[XH * XW * 3];   // 2160 floats = 8.6 KB
    __shared__ float ys[NPAD * 4];      // 2496 floats = 10.0 KB (12 px padding)

    const int tid  = threadIdx.x;
    const int lane = tid & 31;
    const int wave = tid >> 5;
    const int n    = lane & 15;   // N (output channel) for B/C/D; M for A
    const int half = lane >> 4;   // K-pair selector for A/B; M offset for C/D

    const int gx = blockIdx.x * TW;
    const int gy = blockIdx.y * TH;
    const int b  = blockIdx.z;

    // ---- Stage input tile into LDS (read x exactly once from HBM) ----
    const float* xb = x + (size_t)b * 512 * 512 * 3;
    for (int i = tid; i < XH * XW; i += 256) {
        int py = i / XW, px = i - py * XW;
        int sy = min(gy + py, 511);          // clamp at image edge (values unused)
        int sx = min(gx + px, 511);
        const float* src = xb + ((size_t)sy * 512 + sx) * 3;
        xs[i * 3 + 0] = src[0];
        xs[i * 3 + 1] = src[1];
        xs[i * 3 + 2] = src[2];
    }

    // ---- Per-lane B operands for the 9 taps (built once, kept in VGPRs) ----
    // B (4x16 f32): VGPR0 = row K = 2*half, VGPR1 = row K = 1 (half0) / K=3 pad (half1)
    v2f bt[9];
#pragma unroll
    for (int t = 0; t < 9; ++t) {
        float bx = 0.f, by = 0.f;
        if (n < 4) {
            bx = w1[t * 12 + (half ? 8 : 0) + n]; // ic = 2*half, oc = n
            if (!half) by = w1[t * 12 + 4 + n];   // ic = 1
        }
        bt[t].x = bx;
        bt[t].y = by;
    }
    const float bias1 = (n < 4) ? b1[n] : 0.f;

    __syncthreads();

    // ---- Conv1 via V_WMMA_F32_16X16X4_F32: 16 pixels/call, 9 taps accumulated ----
    for (int g = wave; g * 16 < NPIX; g += 8) {
        int p  = g * 16 + n;                 // A: M = lane & 15 (both halves)
        int pc = min(p, NPIX - 1);           // clamp: padded pixels read safely
        int cy = pc / CW, cx = pc - cy * CW;
        v8f acc = {};
#pragma unroll
        for (int ky = 0; ky < 3; ++ky) {
#pragma unroll
            for (int kx = 0; kx < 3; ++kx) {
                int ib = ((cy + ky) * XW + (cx + kx)) * 3;
                // A (16x4 f32): half0 lane -> {ch0, ch1}; half1 lane -> {ch2, 0}
                float a0 = xs[ib + 2 * half];
                float a1 = xs[ib + 1];
                v2f a;
                a.x = a0;
                a.y = half ? 0.f : a1;       // branchless select: EXEC stays all-ones
                acc = __builtin_amdgcn_wmma_f32_16x16x4_f32(
                    false, a, false, bt[ky * 3 + kx], (short)0, acc, false, false);
            }
        }
        // D layout: lane(half,n), VGPR v -> M = v + 8*half, N = n.
        // ys is padded to NPAD pixels, so no per-v bound check -> straight-line
        // stores under a single n<4 exec mask (pairable ds_store_2addr).
        if (n < 4) {
            int base = g * 16 + 8 * half;
#pragma unroll
            for (int v = 0; v < 8; ++v)
                ys[(base + v) * 4 + n] = fmaxf(acc[v] + bias1, 0.f);
        }
    }

    __syncthreads();

    // ---- Conv2 (4 -> 1) + ReLU in VALU; each thread does 2 output pixels ----
    const int lx  = tid & 31;
    const int ly0 = tid >> 5;                // 0..7 ; handles ly0 and ly0+8
    const float bias2 = b2[0];
#pragma unroll
    for (int rep = 0; rep < 2; ++rep) {
        int ly = ly0 + rep * 8;
        float acc = bias2;
#pragma unroll
        for (int t = 0; t < 9; ++t) {
            int ty = t / 3, tx = t - ty * 3;
            const float* yp = &ys[((ly + ty) * CW + (lx + tx)) * 4];
#pragma unroll
            for (int c = 0; c < 4; ++c)
                acc += yp[c] * w2[t * 4 + c];  // w2 uniform -> scalar loads
        }
        int zx = gx + lx, zy = gy + ly;
        if (zx < 508 && zy < 508)
            out[((size_t)b * 508 + zy) * 508 + zx] = fmaxf(acc, 0.f);
    }
}

extern "C" void kernel_launch(void* const* d_in, const int* in_sizes, int n_in,
                              void* d_out, int out_size, void* d_ws, size_t ws_size,
                              hipStream_t stream) {
    (void)in_sizes; (void)n_in; (void)d_ws; (void)ws_size; (void)out_size;
    const float* x  = (const float*)d_in[0];
    const float* w1 = (const float*)d_in[1];
    const float* b1 = (const float*)d_in[2];
    const float* w2 = (const float*)d_in[3];
    const float* b2 = (const float*)d_in[4];
    float* out = (float*)d_out;

    dim3 grid((508 + TW - 1) / TW,   // 16 tiles in W
              (508 + TH - 1) / TH,   // 32 tiles in H
              16);                   // batch
    dim3 block(256);
    fused_conv2x_relu_kernel<<<grid, block, 0, stream>>>(x, w1, b1, w2, b2, out);
}